// CrossPixContrastiveL2_41678362640817
// MI455X (gfx1250) — compile-verified
//
#include <hip/hip_runtime.h>
#include <hip/hip_bf16.h>

typedef float v2f __attribute__((ext_vector_type(2)));
typedef float v8f __attribute__((ext_vector_type(8)));

#define C_DIM 128
#define HW_DIM 1024
#define TEMP_INV 0.1f
#define EPS_F 1e-6f

// ---------------------------------------------------------------------------
// Kernel 1: per-pixel squared norms of rgb / ir features, and zero the
// row/col accumulator arrays (keeps kernel_launch deterministic without
// relying on memset or prior buffer state).
// ws layout (floats): [0]=rnorm2, [NHW]=inorm2, [2*NHW]=rowNum,
// [3*NHW]=rowDen, [4*NHW]=colNum, [5*NHW]=colDen
// ---------------------------------------------------------------------------
__global__ void __launch_bounds__(256)
cpc_norms_zero_kernel(const float* __restrict__ rgb,
                      const float* __restrict__ ir,
                      float* __restrict__ ws, int NHW) {
  int tid = blockIdx.x * blockDim.x + threadIdx.x;
  if (tid >= NHW) return;
  int n = tid >> 10;                 // HW_DIM == 1024
  int p = tid & (HW_DIM - 1);
  const float* a = rgb + (size_t)n * C_DIM * HW_DIM + p;
  const float* b = ir  + (size_t)n * C_DIM * HW_DIM + p;
  float sa = 0.f, sb = 0.f;
#pragma unroll 8
  for (int c = 0; c < C_DIM; ++c) {
    float va = a[c * HW_DIM]; sa += va * va;
    float vb = b[c * HW_DIM]; sb += vb * vb;
  }
  ws[tid]            = sa;   // rnorm2
  ws[NHW + tid]      = sb;   // inorm2
  ws[2 * NHW + tid]  = 0.f;  // rowNum
  ws[3 * NHW + tid]  = 0.f;  // rowDen
  ws[4 * NHW + tid]  = 0.f;  // colNum
  ws[5 * NHW + tid]  = 0.f;  // colDen
}

// ---------------------------------------------------------------------------
// Kernel 2: one wave32 per 16x16 (p,q) tile of the NxHWxHW similarity matrix.
// fp32 WMMA (V_WMMA_F32_16X16X4_F32) accumulates the dot products over C=128,
// then each lane evaluates logit = exp(exp(-dist)/T), applies the label mask,
// reduces rows/cols inside the wave with shuffles, and atomically accumulates
// into the global row/col numerator/denominator arrays.
// ---------------------------------------------------------------------------
__global__ void __launch_bounds__(256)
cpc_tile_kernel(const float* __restrict__ rgb,
                const float* __restrict__ ir,
                const int* __restrict__ rm,
                const int* __restrict__ im,
                float* __restrict__ ws, int NHW) {
  const int lane = threadIdx.x & 31;
  const int tile = blockIdx.x * 8 + (threadIdx.x >> 5);   // 8 waves / block

  // 64 x 64 tiles of 16x16 per image
  const int n  = tile >> 12;
  const int tp = (tile >> 6) & 63;
  const int tq = tile & 63;
  const int p0 = tp << 4;
  const int q0 = tq << 4;

  const int m  = lane & 15;           // M (A) / N (B) index within tile
  const int kb = (lane >> 4) << 1;    // K sub-offset: 0 for lanes 0-15, 2 for 16-31

  const float* Abase = rgb + (size_t)n * C_DIM * HW_DIM + p0 + m;
  const float* Bbase = ir  + (size_t)n * C_DIM * HW_DIM + q0 + m;

  v8f acc = {};
#pragma unroll
  for (int k = 0; k < C_DIM; k += 4) {
    v2f a, b;
    a.x = Abase[(k + kb)     * HW_DIM];
    a.y = Abase[(k + kb + 1) * HW_DIM];
    b.x = Bbase[(k + kb)     * HW_DIM];
    b.y = Bbase[(k + kb + 1) * HW_DIM];
    // D = A(16x4,f32) * B(4x16,f32) + C ; 8-arg form:
    // (neg_a, A, neg_b, B, c_mod, C, reuse_a, reuse_b)
    acc = __builtin_amdgcn_wmma_f32_16x16x4_f32(false, a, false, b,
                                                (short)0, acc, false, false);
  }

  const float* rnorm2 = ws;
  const float* inorm2 = ws + NHW;
  float* rowNum = ws + 2 * NHW;
  float* rowDen = ws + 3 * NHW;
  float* colNum = ws + 4 * NHW;
  float* colDen = ws + 5 * NHW;

  const int q  = q0 + m;
  const int gq = n * HW_DIM + q;
  const float inq = inorm2[gq];
  const int  labq = im[gq];

  const int halfOff = (lane >> 4) << 3;   // C/D layout: lanes 16-31 hold M = j+8
  float colL = 0.f, colM = 0.f;

#pragma unroll
  for (int j = 0; j < 8; ++j) {
    const int p  = p0 + j + halfOff;
    const int gp = n * HW_DIM + p;
    const float rnp = rnorm2[gp];
    const int  labp = rm[gp];

    const float dist  = rnp + inq - 2.0f * acc[j];
    const float logit = expf(expf(-dist) * TEMP_INV);
    const float lm    = (labp == labq) ? logit : 0.f;

    colL += logit;
    colM += lm;

    // row sum: reduce across the 16 lanes of this half-wave (same p, q varies)
    float rl = logit, rlm = lm;
#pragma unroll
    for (int s = 1; s < 16; s <<= 1) {
      rl  += __shfl_xor(rl,  s, 32);
      rlm += __shfl_xor(rlm, s, 32);
    }
    if (m == 0) {
      atomicAdd(&rowNum[gp], rlm);
      atomicAdd(&rowDen[gp], rl);
    }
  }

  // col sums: lanes L and L+16 hold the same q -> fold halves
  colL += __shfl_xor(colL, 16, 32);
  colM += __shfl_xor(colM, 16, 32);
  if (lane < 16) {
    atomicAdd(&colNum[gq], colM);
    atomicAdd(&colDen[gq], colL);
  }
}

// ---------------------------------------------------------------------------
// Kernel 3: masked -log mean over the 2*N*HW loss entries -> scalar.
// ---------------------------------------------------------------------------
__global__ void __launch_bounds__(256)
cpc_finalize_kernel(const int* __restrict__ rm,
                    const int* __restrict__ im,
                    const float* __restrict__ ws,
                    float* __restrict__ out, int NHW) {
  const float* rowNum = ws + 2 * NHW;
  const float* rowDen = ws + 3 * NHW;
  const float* colNum = ws + 4 * NHW;
  const float* colDen = ws + 5 * NHW;

  float total = 0.f, count = 0.f;
  for (int i = threadIdx.x; i < 2 * NHW; i += 256) {
    float num, den; int lab;
    if (i < NHW) { num = rowNum[i]; den = rowDen[i]; lab = rm[i]; }
    else { int j = i - NHW; num = colNum[j]; den = colDen[j]; lab = im[j]; }
    float v = num / (den + EPS_F);
    if (lab > 0 && v != 0.f) { total += -logf(v); count += 1.f; }
  }

  __shared__ float sT[256];
  __shared__ float sC[256];
  sT[threadIdx.x] = total;
  sC[threadIdx.x] = count;
  __syncthreads();
#pragma unroll
  for (int s = 128; s > 0; s >>= 1) {
    if (threadIdx.x < s) {
      sT[threadIdx.x] += sT[threadIdx.x + s];
      sC[threadIdx.x] += sC[threadIdx.x + s];
    }
    __syncthreads();
  }
  if (threadIdx.x == 0) out[0] = sT[0] / sC[0];
}

extern "C" void kernel_launch(void* const* d_in, const int* in_sizes, int n_in,
                              void* d_out, int out_size, void* d_ws, size_t ws_size,
                              hipStream_t stream) {
  const float* rgb = (const float*)d_in[0];
  const float* ir  = (const float*)d_in[1];
  const int*   rm  = (const int*)d_in[2];
  const int*   im  = (const int*)d_in[3];
  float* out = (float*)d_out;
  float* ws  = (float*)d_ws;

  const int NHW = in_sizes[2];          // N * H * W  (N*1024)
  const int N   = NHW / HW_DIM;

  cpc_norms_zero_kernel<<<(NHW + 255) / 256, 256, 0, stream>>>(rgb, ir, ws, NHW);

  const int tiles  = N * 64 * 64;       // 16x16 tiles over (HW x HW) per image
  const int blocks = tiles / 8;         // 8 waves per 256-thread block
  cpc_tile_kernel<<<blocks, 256, 0, stream>>>(rgb, ir, rm, im, ws, NHW);

  cpc_finalize_kernel<<<1, 256, 0, stream>>>(rm, im, ws, out, NHW);
}